// Lstm_29987461660977
// MI455X (gfx1250) — compile-verified
//
#include <hip/hip_runtime.h>
#include <math.h>

#define T_STEPS 10
#define BATCH   64
#define IN      2048
#define HID     2048

typedef __attribute__((ext_vector_type(2))) float v2f;
typedef __attribute__((ext_vector_type(8))) float v8f;

__device__ __forceinline__ float fast_sigmoid(float x) {
    // 1/(1+e^-x) with hardware v_exp/v_rcp (fine for a gate nonlinearity)
    return __builtin_amdgcn_rcpf(1.0f + __expf(-x));
}

// out[M, HID] = A[M, IN] * W[IN, HID]   (MODE==1: sigmoid(acc + bias))
// A row mapping:
//   t_fixed >= 0 : row r is x[r, t_fixed, :]            (M == BATCH)
//   t_fixed <  0 : row r=(t*BATCH+b) is x[b, t, :]      (M == T_STEPS*BATCH)
// One wave computes a 32(M) x 64(N) tile via V_WMMA_F32_16X16X4_F32:
// 8 independent f32 accumulators, each B fragment feeds 2 WMMAs.
template <int MODE>
__global__ void wmma_gemm_f32(const float* __restrict__ X, int t_fixed,
                              const float* __restrict__ W,
                              const float* __restrict__ bias,
                              float* __restrict__ out, int M)
{
    const int lane = threadIdx.x & 31;
    const int half = lane >> 4;      // 0 -> K pair {k,k+1}, 1 -> {k+2,k+3}
    const int lrow = lane & 15;

    const int gwave   = (blockIdx.x * blockDim.x + threadIdx.x) >> 5;
    const int ntilesN = HID / 64;    // 32
    const int m_tile  = gwave / ntilesN;
    const int n_tile  = gwave % ntilesN;
    if (m_tile * 32 >= M) return;    // wave-uniform; EXEC stays all-ones

    // Per-lane A row pointers (A-matrix layout: lanes 0-15 = rows, dup 16-31)
    const float* arow0;
    const float* arow1;
    {
        const int mrow0 = m_tile * 32 + lrow;
        const int mrow1 = mrow0 + 16;
        if (t_fixed >= 0) {
            arow0 = X + (size_t)mrow0 * (T_STEPS * IN) + (size_t)t_fixed * IN;
            arow1 = X + (size_t)mrow1 * (T_STEPS * IN) + (size_t)t_fixed * IN;
        } else {
            const int t0 = mrow0 / BATCH, b0 = mrow0 - t0 * BATCH;
            const int t1 = mrow1 / BATCH, b1 = mrow1 - t1 * BATCH;
            arow0 = X + (size_t)b0 * (T_STEPS * IN) + (size_t)t0 * IN;
            arow1 = X + (size_t)b1 * (T_STEPS * IN) + (size_t)t1 * IN;
        }
    }

    const int n0 = n_tile * 64;

    v8f acc0[4] = {v8f{}, v8f{}, v8f{}, v8f{}};
    v8f acc1[4] = {v8f{}, v8f{}, v8f{}, v8f{}};

    #pragma unroll 2
    for (int k = 0; k < IN; k += 4) {
        const int ka = k + half * 2;
        // A 16x4 tiles: lane holds A[mrow][ka], A[mrow][ka+1]  (8B loads)
        v2f a0 = *(const v2f*)(arow0 + ka);
        v2f a1 = *(const v2f*)(arow1 + ka);
        // B 4x16 tiles: VGPR0 = K row (ka), VGPR1 = K row (ka+1), col = lrow
        const float* w0 = W + (size_t)ka * HID + n0 + lrow;
        const float* w1 = w0 + HID;
        #pragma unroll
        for (int j = 0; j < 4; ++j) {
            v2f b;
            b[0] = w0[j * 16];
            b[1] = w1[j * 16];
            acc0[j] = __builtin_amdgcn_wmma_f32_16x16x4_f32(
                false, a0, false, b, (short)0, acc0[j], false, false);
            acc1[j] = __builtin_amdgcn_wmma_f32_16x16x4_f32(
                false, a1, false, b, (short)0, acc1[j], false, false);
        }
    }

    // C/D layout: VGPR r -> row (base + half*8 + r), col = n0 + j*16 + lrow
    #pragma unroll
    for (int j = 0; j < 4; ++j) {
        const int ncol = n0 + j * 16 + lrow;
        const float bv = (MODE == 1) ? bias[ncol] : 0.0f;
        #pragma unroll
        for (int r = 0; r < 8; ++r) {
            const int row0 = m_tile * 32 + half * 8 + r;
            float v0 = acc0[j][r];
            float v1 = acc1[j][r];
            if (MODE == 1) {
                v0 = fast_sigmoid(v0 + bv);
                v1 = fast_sigmoid(v1 + bv);
            }
            out[(size_t)row0 * HID + ncol]        = v0;
            out[(size_t)(row0 + 16) * HID + ncol] = v1;
        }
    }
}

// Sequential T=10 recurrence per (b,h) column. reverse selects time order of
// the inputs (backward cell consumes xt[::-1]); outputs written in scan order.
__global__ void scan_kernel(const float* __restrict__ xs,
                            const float* __restrict__ fs,
                            float* __restrict__ ss,
                            float* __restrict__ ff,
                            int reverse)
{
    const int idx = blockIdx.x * blockDim.x + threadIdx.x;   // b*HID + h
    if (idx >= BATCH * HID) return;
    float s = 0.0f, f = 0.0f;
    #pragma unroll
    for (int t = 0; t < T_STEPS; ++t) {
        const int tsrc = reverse ? (T_STEPS - 1 - t) : t;
        const float xv = xs[(size_t)tsrc * BATCH * HID + idx];
        const float fv = fs[(size_t)tsrc * BATCH * HID + idx];
        s = xv * (1.0f - fv) + s * fv;
        f = xv * fv + f * (1.0f - fv);
        ss[(size_t)t * BATCH * HID + idx] = s;
        ff[(size_t)t * BATCH * HID + idx] = f;
    }
}

// h = tanh(s_T)*r_s + (1-r_s)*xr + tanh(f_T)*r_f + (1-r_f)*xr
__global__ void head_kernel(const float* __restrict__ ss,
                            const float* __restrict__ ff,
                            const float* __restrict__ xr,
                            const float* __restrict__ rs,
                            const float* __restrict__ rf,
                            float* __restrict__ prev_x,
                            int dir)
{
    const int idx = blockIdx.x * blockDim.x + threadIdx.x;   // b*HID + h
    if (idx >= BATCH * HID) return;
    const int b = idx / HID;
    const int h = idx - b * HID;
    const float s  = ss[(size_t)(T_STEPS - 1) * BATCH * HID + idx];
    const float f  = ff[(size_t)(T_STEPS - 1) * BATCH * HID + idx];
    const float r1 = rs[idx];
    const float r2 = rf[idx];
    const float x  = xr[idx];
    const float hv = tanhf(s) * r1 + (1.0f - r1) * x
                   + tanhf(f) * r2 + (1.0f - r2) * x;
    prev_x[(size_t)b * (2 * HID) + (size_t)dir * HID + h] = hv;
}

extern "C" void kernel_launch(void* const* d_in, const int* in_sizes, int n_in,
                              void* d_out, int out_size, void* d_ws, size_t ws_size,
                              hipStream_t stream)
{
    const float* x      = (const float*)d_in[0];
    const float* f_w_s  = (const float*)d_in[1];
    const float* f_wx   = (const float*)d_in[2];
    const float* f_wf_s = (const float*)d_in[3];
    const float* f_wr_s = (const float*)d_in[4];
    const float* f_wr_f = (const float*)d_in[5];
    const float* f_bf_s = (const float*)d_in[6];
    const float* f_br_s = (const float*)d_in[7];
    const float* f_br_f = (const float*)d_in[8];
    const float* b_w_s  = (const float*)d_in[9];
    const float* b_wx   = (const float*)d_in[10];
    const float* b_wf_s = (const float*)d_in[11];
    const float* b_wr_s = (const float*)d_in[12];
    const float* b_wr_f = (const float*)d_in[13];
    const float* b_bf_s = (const float*)d_in[14];
    const float* b_br_s = (const float*)d_in[15];
    const float* b_br_f = (const float*)d_in[16];

    float* out = (float*)d_out;

    // Output layout: [prev_x (B*2H) | ss_f | ff_f | ss_b | ff_b] (each T*B*H)
    const size_t TBH = (size_t)T_STEPS * BATCH * HID;
    const size_t BH  = (size_t)BATCH * HID;
    float* out_ss_f = out + (size_t)BATCH * 2 * HID;
    float* out_ff_f = out_ss_f + TBH;
    float* out_ss_b = out_ff_f + TBH;
    float* out_ff_b = out_ss_b + TBH;

    // Workspace: 4*TBH + 6*BH floats (~24 MB)
    float* ws   = (float*)d_ws;
    float* xs_f = ws; ws += TBH;
    float* fs_f = ws; ws += TBH;
    float* xs_b = ws; ws += TBH;
    float* fs_b = ws; ws += TBH;
    float* xr_f = ws; ws += BH;
    float* rs_f = ws; ws += BH;
    float* rf_f = ws; ws += BH;
    float* xr_b = ws; ws += BH;
    float* rs_b = ws; ws += BH;
    float* rf_b = ws; ws += BH;

    const int ntilesN = HID / 64;
    auto gemm = [&](int t_fixed, const float* W, const float* bias,
                    float* o, int M, int mode) {
        const int waves   = (M / 32) * ntilesN;
        const int threads = 256;                 // 8 waves per block (wave32)
        const int blocks  = (waves * 32) / threads;
        if (mode)
            wmma_gemm_f32<1><<<blocks, threads, 0, stream>>>(x, t_fixed, W, bias, o, M);
        else
            wmma_gemm_f32<0><<<blocks, threads, 0, stream>>>(x, t_fixed, W, bias, o, M);
    };

    const int MT = T_STEPS * BATCH;              // 640 rows

    // Big GEMMs (both directions run in forward time order; reversal in scan)
    gemm(-1, f_w_s,  nullptr, xs_f, MT, 0);
    gemm(-1, f_wf_s, f_bf_s,  fs_f, MT, 1);
    gemm(-1, b_w_s,  nullptr, xs_b, MT, 0);
    gemm(-1, b_wf_s, b_bf_s,  fs_b, MT, 1);

    const int nscan = BATCH * HID;
    scan_kernel<<<nscan / 256, 256, 0, stream>>>(xs_f, fs_f, out_ss_f, out_ff_f, 0);
    scan_kernel<<<nscan / 256, 256, 0, stream>>>(xs_b, fs_b, out_ss_b, out_ff_b, 1);

    // Head GEMMs on the last consumed timestep of each direction
    gemm(T_STEPS - 1, f_wx,   nullptr, xr_f, BATCH, 0);
    gemm(T_STEPS - 1, f_wr_s, f_br_s,  rs_f, BATCH, 1);
    gemm(T_STEPS - 1, f_wr_f, f_br_f,  rf_f, BATCH, 1);
    gemm(0,           b_wx,   nullptr, xr_b, BATCH, 0);
    gemm(0,           b_wr_s, b_br_s,  rs_b, BATCH, 1);
    gemm(0,           b_wr_f, b_br_f,  rf_b, BATCH, 1);

    head_kernel<<<nscan / 256, 256, 0, stream>>>(out_ss_f, out_ff_f, xr_f, rs_f, rf_f, out, 0);
    head_kernel<<<nscan / 256, 256, 0, stream>>>(out_ss_b, out_ff_b, xr_b, rs_b, rf_b, out, 1);
}